// Swin3D_50766513438995
// MI455X (gfx1250) — compile-verified
//
#include <hip/hip_runtime.h>

#define B_  8
#define N_  8192
#define H_  256
#define NUP 2048      // N/4
#define NDN 6144      // N - NUP
#define KM  5
#define ENC_NEGINF 0x007fffffu

typedef float v2f __attribute__((ext_vector_type(2)));
typedef float v8f __attribute__((ext_vector_type(8)));

// monotonic float<->uint order-preserving encoding (for atomic max on floats)
__device__ __forceinline__ unsigned enc_f32(float f) {
    unsigned u = __float_as_uint(f);
    return (u & 0x80000000u) ? ~u : (u | 0x80000000u);
}
__device__ __forceinline__ float dec_f32(unsigned e) {
    unsigned u = (e & 0x80000000u) ? (e & 0x7fffffffu) : ~e;
    return __uint_as_float(u);
}

// ---------------- init: encoded -inf into output accumulator, zero counters --------
__global__ void k_init(unsigned* __restrict__ enc_out, int* __restrict__ counters) {
    size_t total  = (size_t)B_ * N_ * H_;
    size_t stride = (size_t)gridDim.x * blockDim.x;
    for (size_t i = (size_t)blockIdx.x * blockDim.x + threadIdx.x; i < total; i += stride)
        enc_out[i] = ENC_NEGINF;
    if (blockIdx.x == 0 && threadIdx.x < 2 * B_) counters[threadIdx.x] = 0;
}

// ---------------- top-25% selection via brute-force rank against LDS scores -------
__global__ void k_select(const float* __restrict__ scores, float* __restrict__ mask_out,
                         int* __restrict__ up_list, int* __restrict__ dn_list,
                         int* __restrict__ counters) {
    __shared__ float s_sc[N_];                 // 32 KB
    int b   = blockIdx.x >> 5;                 // 32 blocks per batch
    int blk = blockIdx.x & 31;
    const float* sc = scores + (size_t)b * N_;
    for (int j = threadIdx.x; j < N_; j += 256) s_sc[j] = sc[j];
    __syncthreads();

    int   i   = blk * 256 + (int)threadIdx.x;
    float mys = s_sc[i];
    int rank = 0;
    #pragma unroll 8
    for (int j = 0; j < N_; ++j) {
        float v = s_sc[j];
        // stable argsort(-scores): strictly greater, or equal with smaller index
        rank += (v > mys) || (v == mys && j < i);
    }
    bool up = rank < NUP;
    mask_out[(size_t)b * N_ + i] = up ? 1.0f : 0.0f;
    if (up) { int p = atomicAdd(&counters[2 * b],     1); up_list[b * NUP + p] = i; }
    else    { int p = atomicAdd(&counters[2 * b + 1], 1); dn_list[b * NDN + p] = i; }
}

// ---------------- feat = h @ W[0:256] (+ s_l, scores rank-4 epilogue) + bias -------
// block = 256 threads = 8 waves; block covers one 16-row tile x all 256 cols.
// wave w computes col tiles w and w+8 (A fragment reused across both WMMAs).
__global__ __launch_bounds__(256)
void k_gemm(const float* __restrict__ h, const float* __restrict__ s_l,
            const float* __restrict__ scores, const float* __restrict__ W,
            const float* __restrict__ bias, float* __restrict__ feat) {
    int b    = blockIdx.x >> 9;                // 512 row tiles per batch
    int rt   = blockIdx.x & 511;
    int row0 = rt * 16;
    int wave = threadIdx.x >> 5;
    int lane = threadIdx.x & 31;
    int half = lane >> 4;                      // lanes 16-31 hold K+2/K+3 of A
    int l15  = lane & 15;

    const float* X = h + ((size_t)b * N_ + row0) * H_;   // 16 x 256 tile, row stride H_
    int col0a = wave * 16;
    int col0b = (wave + 8) * 16;

    v8f acc0 = {}, acc1 = {};
    for (int k0 = 0; k0 < H_; k0 += 4) {
        int ka = k0 + 2 * half;
        v2f a, b0, b1;
        // A 16x4: lane l15 = row, VGPR0/1 = K = ka, ka+1
        a.x = X[(size_t)l15 * H_ + ka];
        a.y = X[(size_t)l15 * H_ + ka + 1];
        // B 4x16: lane l15 = col, VGPR0/1 = K = ka, ka+1 (halves split K like A)
        b0.x = W[(size_t)ka * H_ + col0a + l15];
        b0.y = W[(size_t)(ka + 1) * H_ + col0a + l15];
        b1.x = W[(size_t)ka * H_ + col0b + l15];
        b1.y = W[(size_t)(ka + 1) * H_ + col0b + l15];
        acc0 = __builtin_amdgcn_wmma_f32_16x16x4_f32(false, a, false, b0, (short)0, acc0, false, false);
        acc1 = __builtin_amdgcn_wmma_f32_16x16x4_f32(false, a, false, b1, (short)0, acc1, false, false);
    }

    // epilogue: += s_l @ W[256:259] + scores * W[259] + bias
    #pragma unroll
    for (int t = 0; t < 2; ++t) {
        int col = (t ? col0b : col0a) + l15;
        v8f c   = t ? acc1 : acc0;
        float w0 = W[(size_t)256 * H_ + col];
        float w1 = W[(size_t)257 * H_ + col];
        float w2 = W[(size_t)258 * H_ + col];
        float w3 = W[(size_t)259 * H_ + col];
        float bb = bias[col];
        #pragma unroll
        for (int v = 0; v < 8; ++v) {
            int row = row0 + v + 8 * half;      // C/D layout: VGPR v, halves offset by 8
            const float* sl = s_l + ((size_t)b * N_ + row) * 3;
            float scv = scores[(size_t)b * N_ + row];
            float val = c[v] + sl[0] * w0 + sl[1] * w1 + sl[2] * w2 + scv * w3 + bb;
            feat[((size_t)b * N_ + row) * H_ + col] = val;
        }
    }
}

// ---------------- brute-force 5-NN of each down node against up set (LDS) ----------
__global__ void k_knn(const float* __restrict__ s_l, const int* __restrict__ up_list,
                      const int* __restrict__ dn_list, int* __restrict__ nbr) {
    __shared__ float ux[NUP], uy[NUP], uz[NUP];  // 24 KB
    __shared__ int   uid[NUP];                   //  8 KB
    int b   = blockIdx.x / 24;                   // 24 blocks per batch (6144/256)
    int blk = blockIdx.x % 24;
    for (int j = threadIdx.x; j < NUP; j += 256) {
        int u = up_list[b * NUP + j];
        const float* p = s_l + ((size_t)b * N_ + u) * 3;
        ux[j] = p[0]; uy[j] = p[1]; uz[j] = p[2]; uid[j] = u;
    }
    __syncthreads();

    int dl = blk * 256 + (int)threadIdx.x;
    int d  = dn_list[b * NDN + dl];
    const float* p = s_l + ((size_t)b * N_ + d) * 3;
    float px = p[0], py = p[1], pz = p[2];

    float bd0 = 3.4e38f, bd1 = 3.4e38f, bd2 = 3.4e38f, bd3 = 3.4e38f, bd4 = 3.4e38f;
    int   bi0 = 0, bi1 = 0, bi2 = 0, bi3 = 0, bi4 = 0;
    for (int j = 0; j < NUP; ++j) {
        float dx = ux[j] - px, dy = uy[j] - py, dz = uz[j] - pz;
        float d2 = dx * dx + dy * dy + dz * dz;
        if (d2 < bd4) {
            int u = uid[j];
            if (d2 < bd0)      { bd4=bd3;bi4=bi3; bd3=bd2;bi3=bi2; bd2=bd1;bi2=bi1; bd1=bd0;bi1=bi0; bd0=d2;bi0=u; }
            else if (d2 < bd1) { bd4=bd3;bi4=bi3; bd3=bd2;bi3=bi2; bd2=bd1;bi2=bi1; bd1=d2;bi1=u; }
            else if (d2 < bd2) { bd4=bd3;bi4=bi3; bd3=bd2;bi3=bi2; bd2=d2;bi2=u; }
            else if (d2 < bd3) { bd4=bd3;bi4=bi3; bd3=d2;bi3=u; }
            else               { bd4=d2;bi4=u; }
        }
    }
    int base = (b * NDN + dl) * KM;
    nbr[base + 0] = bi0; nbr[base + 1] = bi1; nbr[base + 2] = bi2;
    nbr[base + 3] = bi3; nbr[base + 4] = bi4;
}

// ---------------- scatter max: feat[down] -> 5 up rows via encoded atomicMax -------
__global__ void k_scatter(const float* __restrict__ feat, const int* __restrict__ dn_list,
                          const int* __restrict__ nbr, unsigned* __restrict__ enc_out) {
    int e = blockIdx.x;                 // b*NDN + dl
    int b = e / NDN;
    int t = threadIdx.x;
    int d = dn_list[e];
    float f = feat[((size_t)b * N_ + d) * H_ + t];
    unsigned enc = enc_f32(f);
    const int* nb = nbr + (size_t)e * KM;
    #pragma unroll
    for (int m = 0; m < KM; ++m) {
        int u = nb[m];
        atomicMax(&enc_out[((size_t)b * N_ + u) * H_ + t], enc);
    }
}

// ---------------- decode in place; untouched rows (-inf) become zero ---------------
__global__ void k_finalize(unsigned* __restrict__ enc_out, float* __restrict__ out) {
    size_t total  = (size_t)B_ * N_ * H_;
    size_t stride = (size_t)gridDim.x * blockDim.x;
    for (size_t i = (size_t)blockIdx.x * blockDim.x + threadIdx.x; i < total; i += stride) {
        unsigned e = enc_out[i];
        out[i] = (e == ENC_NEGINF) ? 0.0f : dec_f32(e);
    }
}

extern "C" void kernel_launch(void* const* d_in, const int* in_sizes, int n_in,
                              void* d_out, int out_size, void* d_ws, size_t ws_size,
                              hipStream_t stream) {
    const float* h      = (const float*)d_in[0];   // [B,N,H]
    const float* s_l    = (const float*)d_in[1];   // [B,N,3]
    const float* scores = (const float*)d_in[2];   // [B,N]
    const float* W_emb  = (const float*)d_in[3];   // [260,256]
    const float* b_emb  = (const float*)d_in[4];   // [256]

    // workspace layout
    char*  ws       = (char*)d_ws;
    float* feat     = (float*)ws;                                   // B*N*H f32 (64 MB)
    size_t featB    = (size_t)B_ * N_ * H_ * sizeof(float);
    int*   up_list  = (int*)(ws + featB);                           // B*NUP
    int*   dn_list  = up_list + (size_t)B_ * NUP;                   // B*NDN
    int*   nbr      = dn_list + (size_t)B_ * NDN;                   // B*NDN*KM
    int*   counters = nbr + (size_t)B_ * NDN * KM;                  // 2*B

    unsigned* enc_out  = (unsigned*)d_out;                          // aliases out floats
    float*    out_f    = (float*)d_out;
    float*    mask_out = out_f + (size_t)B_ * N_ * H_;              // B*N mask floats

    k_init    <<<4096,        256, 0, stream>>>(enc_out, counters);
    k_select  <<<B_ * 32,     256, 0, stream>>>(scores, mask_out, up_list, dn_list, counters);
    k_gemm    <<<B_ * 512,    256, 0, stream>>>(h, s_l, scores, W_emb, b_emb, feat);
    k_knn     <<<B_ * 24,     256, 0, stream>>>(s_l, up_list, dn_list, nbr);
    k_scatter <<<B_ * NDN,    256, 0, stream>>>(feat, dn_list, nbr, enc_out);
    k_finalize<<<4096,        256, 0, stream>>>(enc_out, out_f);
}